// GraphSAGE_4672924418191
// MI455X (gfx1250) — compile-verified
//
#include <hip/hip_runtime.h>
#include <hip/hip_bf16.h>

typedef __attribute__((ext_vector_type(2))) float v2f;
typedef __attribute__((ext_vector_type(8))) float v8f;

#define N_NODES 50000
#define DH 128
#define LDS_STRIDE 132   // 128 + 4 pad: breaks 64-bank conflict on column reads

// ---------------------------------------------------------------- zero scratch
__global__ void zero_ws(float* __restrict__ agg, float* __restrict__ cnt) {
    int i = blockIdx.x * blockDim.x + threadIdx.x;
    if (i < N_NODES * DH) agg[i] = 0.0f;
    if (i < N_NODES)      cnt[i] = 0.0f;
}

// ------------------------------------------------- edge scatter (mean accum)
// one wave32 per edge; lane handles 4 consecutive features (float4 gather,
// 4x global_atomic_add_f32 scatter -- all L2-resident on MI455X)
__global__ __launch_bounds__(256)
void scatter_mean_accum(const float* __restrict__ xin,
                        const long long* __restrict__ src,
                        const long long* __restrict__ dst,
                        float* __restrict__ agg,
                        float* __restrict__ cnt,
                        int nEdges) {
    int wave = threadIdx.x >> 5;
    int lane = threadIdx.x & 31;
    int e = blockIdx.x * (blockDim.x >> 5) + wave;
    if (e >= nEdges) return;
    long long s = src[e];
    long long d = dst[e];
    const float4 v = *(const float4*)(xin + (size_t)s * DH + lane * 4);
    float* a = agg + (size_t)d * DH + lane * 4;
    atomicAdd(a + 0, v.x);
    atomicAdd(a + 1, v.y);
    atomicAdd(a + 2, v.z);
    atomicAdd(a + 3, v.w);
    if (lane == 0) atomicAdd(cnt + d, 1.0f);
}

// ------------------------- fused: mean-div + dual GEMM (WMMA f32) + LN + ReLU
// block = 256 threads = 8 waves; block owns 16 rows x 128 cols of output.
// Each wave computes one 16x16 tile with V_WMMA_F32_16X16X4_F32 over K=256
// (concatenated [mean | x] panel staged in LDS).
__global__ __launch_bounds__(256)
void sage_gemm_ln_relu(const float* __restrict__ xin,
                       const float* __restrict__ agg,
                       const float* __restrict__ cnt,
                       const float* __restrict__ Wl,    // [128,128] row-major (k, n)
                       const float* __restrict__ Wr,    // [128,128]
                       const float* __restrict__ bias,  // [128]
                       const float* __restrict__ gamma, // [128]
                       const float* __restrict__ beta,  // [128]
                       float* __restrict__ hout) {
    __shared__ float meanLds[16 * LDS_STRIDE];
    __shared__ float xLds[16 * LDS_STRIDE];
    __shared__ float outLds[16 * LDS_STRIDE];

    const int m0  = blockIdx.x * 16;
    const int tid = threadIdx.x;

    // ---- stage A-panels (mean & self), fold 1/max(cnt,1) into the copy
    {
        int row = tid >> 4;          // 0..15
        int cb  = (tid & 15) * 8;    // 8 floats per thread per panel
        int n   = m0 + row;
        float inv = 1.0f / fmaxf(cnt[n], 1.0f);
        const float* ar = agg + (size_t)n * DH + cb;
        const float* xr = xin + (size_t)n * DH + cb;
        float* ml = meanLds + row * LDS_STRIDE + cb;
        float* xl = xLds    + row * LDS_STRIDE + cb;
#pragma unroll
        for (int i = 0; i < 8; ++i) {
            ml[i] = ar[i] * inv;
            xl[i] = xr[i];
        }
    }
    __syncthreads();

    const int wave = tid >> 5;     // 0..7 -> column tile
    const int lane = tid & 31;
    const int lr   = lane & 15;
    const int hi   = lane >> 4;    // 0: K=0,1  1: K=2,3 (A/B f32 16x16x4 layout)
    const int n0   = wave * 16;

    v8f c = {0.f, 0.f, 0.f, 0.f, 0.f, 0.f, 0.f, 0.f};

    // mean @ Wl
#pragma unroll 4
    for (int k = 0; k < DH; k += 4) {
        int ka = k + 2 * hi;
        v2f a = *(const v2f*)(meanLds + lr * LDS_STRIDE + ka);
        v2f b;
        b.x = Wl[ka * DH + n0 + lr];
        b.y = Wl[(ka + 1) * DH + n0 + lr];
        c = __builtin_amdgcn_wmma_f32_16x16x4_f32(false, a, false, b,
                                                  (short)0, c, false, false);
    }
    // x @ Wr
#pragma unroll 4
    for (int k = 0; k < DH; k += 4) {
        int ka = k + 2 * hi;
        v2f a = *(const v2f*)(xLds + lr * LDS_STRIDE + ka);
        v2f b;
        b.x = Wr[ka * DH + n0 + lr];
        b.y = Wr[(ka + 1) * DH + n0 + lr];
        c = __builtin_amdgcn_wmma_f32_16x16x4_f32(false, a, false, b,
                                                  (short)0, c, false, false);
    }

    // ---- accumulator + bias -> LDS tile (lane lr = col n0+lr, vgpr v = row v+8*hi)
    {
        int col = n0 + lr;
        float bv = bias[col];
#pragma unroll
        for (int v = 0; v < 8; ++v) {
            outLds[(v + 8 * hi) * LDS_STRIDE + col] = c[v] + bv;
        }
    }
    __syncthreads();

    // ---- LayerNorm + ReLU: 16 threads per row (width-16 shuffle groups align
    //      with lane subgroups), 8 cols per thread.
    {
        int row = tid >> 4;
        int sub = tid & 15;
        const float* rp = outLds + row * LDS_STRIDE + sub * 8;
        float v[8];
        float s = 0.f, ss = 0.f;
#pragma unroll
        for (int i = 0; i < 8; ++i) {
            v[i] = rp[i];
            s  += v[i];
            ss += v[i] * v[i];
        }
#pragma unroll
        for (int m = 8; m >= 1; m >>= 1) {
            s  += __shfl_xor(s, m, 16);
            ss += __shfl_xor(ss, m, 16);
        }
        float mu  = s * (1.0f / DH);
        float var = ss * (1.0f / DH) - mu * mu;
        float rs  = rsqrtf(var + 1e-5f);
        int n = m0 + row;
        float* op = hout + (size_t)n * DH + sub * 8;
#pragma unroll
        for (int i = 0; i < 8; ++i) {
            int col = sub * 8 + i;
            float z = (v[i] - mu) * rs * gamma[col] + beta[col];
            op[i] = fmaxf(z, 0.0f);
        }
    }
}

// ------------------------------------------------ final 128 -> 2 head
// one wave32 per node; lane-strided K, wave reduction.
__global__ __launch_bounds__(256)
void sage_out(const float* __restrict__ hin,
              const float* __restrict__ agg,
              const float* __restrict__ cnt,
              const float* __restrict__ W3l,  // [128,2]
              const float* __restrict__ W3r,  // [128,2]
              const float* __restrict__ b3,   // [2]
              float* __restrict__ out) {
    int wave = threadIdx.x >> 5;
    int lane = threadIdx.x & 31;
    int n = blockIdx.x * (blockDim.x >> 5) + wave;
    if (n >= N_NODES) return;
    float inv = 1.0f / fmaxf(cnt[n], 1.0f);
    float a0 = 0.f, a1 = 0.f;
#pragma unroll
    for (int k = lane; k < DH; k += 32) {
        float m = agg[(size_t)n * DH + k] * inv;
        float h = hin[(size_t)n * DH + k];
        a0 += m * W3l[k * 2 + 0] + h * W3r[k * 2 + 0];
        a1 += m * W3l[k * 2 + 1] + h * W3r[k * 2 + 1];
    }
#pragma unroll
    for (int msk = 16; msk >= 1; msk >>= 1) {
        a0 += __shfl_xor(a0, msk, 32);
        a1 += __shfl_xor(a1, msk, 32);
    }
    if (lane == 0) {
        out[n * 2 + 0] = a0 + b3[0];
        out[n * 2 + 1] = a1 + b3[1];
    }
}

// ----------------------------------------------------------------- launcher
extern "C" void kernel_launch(void* const* d_in, const int* in_sizes, int n_in,
                              void* d_out, int out_size, void* d_ws, size_t ws_size,
                              hipStream_t stream) {
    const float*     x   = (const float*)d_in[0];
    const long long* ei  = (const long long*)d_in[1];
    const int        nE  = in_sizes[1] / 2;
    const long long* src = ei;
    const long long* dst = ei + nE;
    const float *W1l = (const float*)d_in[2],  *W1r = (const float*)d_in[3];
    const float *b1  = (const float*)d_in[4];
    const float *g1  = (const float*)d_in[5],  *be1 = (const float*)d_in[6];
    const float *W2l = (const float*)d_in[7],  *W2r = (const float*)d_in[8];
    const float *b2  = (const float*)d_in[9];
    const float *g2  = (const float*)d_in[10], *be2 = (const float*)d_in[11];
    const float *W3l = (const float*)d_in[12], *W3r = (const float*)d_in[13];
    const float *b3  = (const float*)d_in[14];

    char* ws = (char*)d_ws;
    float* agg = (float*)ws;                                          // 25.6 MB
    float* cnt = (float*)(ws + (size_t)N_NODES * DH * sizeof(float)); // 0.2 MB
    float* hA  = (float*)((char*)cnt + (size_t)N_NODES * sizeof(float));
    float* hB  = hA + (size_t)N_NODES * DH;

    const dim3 blk(256);
    const int zeroBlocks = (N_NODES * DH + 255) / 256;
    const int scatBlocks = (nE + 7) / 8;          // 8 edges (waves) per block
    const int gemmBlocks = N_NODES / 16;          // 3125, exact
    const int outBlocks  = (N_NODES + 7) / 8;

    // layer 1
    zero_ws<<<zeroBlocks, blk, 0, stream>>>(agg, cnt);
    scatter_mean_accum<<<scatBlocks, blk, 0, stream>>>(x, src, dst, agg, cnt, nE);
    sage_gemm_ln_relu<<<gemmBlocks, blk, 0, stream>>>(x, agg, cnt, W1l, W1r, b1, g1, be1, hA);
    // layer 2
    zero_ws<<<zeroBlocks, blk, 0, stream>>>(agg, cnt);
    scatter_mean_accum<<<scatBlocks, blk, 0, stream>>>(hA, src, dst, agg, cnt, nE);
    sage_gemm_ln_relu<<<gemmBlocks, blk, 0, stream>>>(hA, agg, cnt, W2l, W2r, b2, g2, be2, hB);
    // layer 3
    zero_ws<<<zeroBlocks, blk, 0, stream>>>(agg, cnt);
    scatter_mean_accum<<<scatBlocks, blk, 0, stream>>>(hB, src, dst, agg, cnt, nE);
    sage_out<<<outBlocks, blk, 0, stream>>>(hB, agg, cnt, W3l, W3r, b3, (float*)d_out);
}